// LE_ACE_36739150250616
// MI455X (gfx1250) — compile-verified
//
#include <hip/hip_runtime.h>
#include <hip/hip_bf16.h>

typedef __attribute__((ext_vector_type(2))) float v2f;
typedef __attribute__((ext_vector_type(8))) float v8f;

#define N_STRUCT 256
#define N_SPEC   4
#define NSEG     (N_STRUCT * N_SPEC)
#define F2       1174
#define ROWLEN   (N_SPEC + N_SPEC*64 + N_SPEC*F2)   // 4 + 256 + 4696 = 4956
#define B1_OFF   N_SPEC                              // 4
#define B2_OFF   (N_SPEC + N_SPEC*64)                // 260

// ---------------- segment binning (deterministic, no atomics) ----------------

__global__ void seg_count_kernel(const int* __restrict__ st, const int* __restrict__ sp,
                                 int* __restrict__ cnt, int n_atoms) {
    int seg = blockIdx.x * blockDim.x + threadIdx.x;
    if (seg >= NSEG) return;
    int c = 0;
    for (int i = 0; i < n_atoms; ++i)
        c += ((st[i] * N_SPEC + sp[i]) == seg) ? 1 : 0;
    cnt[seg] = c;
}

__global__ void seg_scan_kernel(const int* __restrict__ cnt, int* __restrict__ offs) {
    __shared__ int sm[NSEG];
    int t = threadIdx.x;
    int v = cnt[t];
    sm[t] = v;
    __syncthreads();
    for (int off = 1; off < NSEG; off <<= 1) {
        int x = (t >= off) ? sm[t - off] : 0;
        __syncthreads();
        sm[t] += x;
        __syncthreads();
    }
    offs[t] = sm[t] - v;   // exclusive prefix
}

__global__ void seg_fill_kernel(const int* __restrict__ st, const int* __restrict__ sp,
                                const int* __restrict__ offs, int* __restrict__ list,
                                int n_atoms) {
    int seg = blockIdx.x * blockDim.x + threadIdx.x;
    if (seg >= NSEG) return;
    int pos = offs[seg];
    for (int i = 0; i < n_atoms; ++i)
        if ((st[i] * N_SPEC + sp[i]) == seg) list[pos++] = i;   // index order -> deterministic sums
}

// ---------------- per-segment pair-invariant GEMM via WMMA f32 16x16x4 ----------------
//
// G[q,p] = sum_{a in seg} sum_m  A[q,m,a] * ( sign(m)/sqrt(2l+1) * A[p,d-1-m,a] )
// K re-flattened as (group of 4 atoms) x m : each 16x16x4 chunk has uniform m,
// so sign/rev are compile-time and atom ids are 4 uniform scalars per group.
// B operand of chunk m == sign(m) * (A operand of chunk d-1-m): loads halved.
// A operand (16x4 f32): lane = row + 16*(K>=2), vgpr j holds K = j + 2*(lane>=16).
// C/D (16x16 f32): lane L, reg r -> (M = r + 8*(L>=16), N = L%16).

template <int L, bool TAIL>
__device__ __forceinline__ void mma_group(const float* __restrict__ S, int n_atoms,
                                          int row, int rowh, int at0, int at1,
                                          float w0, float w1,
                                          v8f& c00, v8f& c01, v8f& c11) {
    constexpr int q = (L == 0) ? 32 : (L == 1) ? 24 : (L == 2) ? 20 : 16;
    constexpr int d = 2 * L + 1;
    constexpr float inv = (L == 0) ? 1.0f
                        : (L == 1) ? 0.57735026918962576f   // 1/sqrt(3)
                        : (L == 2) ? 0.44721359549995794f   // 1/sqrt(5)
                                   : 0.37796447300922722f;  // 1/sqrt(7)
    constexpr int DH = (q > 16) ? d : 1;

    float alo[d][2];
    float ahi[DH][2];
#pragma unroll
    for (int m = 0; m < d; ++m) {                     // branch-free gather of all K chunks
        int off = (row * d + m) * n_atoms;
        alo[m][0] = S[off + at0];
        alo[m][1] = S[off + at1];
        if (TAIL) { alo[m][0] *= w0; alo[m][1] *= w1; }
        if constexpr (q > 16) {
            int offh = (rowh * d + m) * n_atoms;      // rowh clamped: padded rows never stored
            ahi[m][0] = S[offh + at0];
            ahi[m][1] = S[offh + at1];
            if (TAIL) { ahi[m][0] *= w0; ahi[m][1] *= w1; }
        }
    }
#pragma unroll
    for (int m = 0; m < d; ++m) {
        constexpr int dm1 = d - 1;
        const int mr = dm1 - m;
        const float sg = (m & 1) ? -inv : inv;        // literal per unrolled m
        v2f A0 = {alo[m][0], alo[m][1]};
        v2f B0 = {sg * alo[mr][0], sg * alo[mr][1]};
        c00 = __builtin_amdgcn_wmma_f32_16x16x4_f32(false, A0, false, B0, (short)0, c00, false, false);
        if constexpr (q > 16) {
            v2f A1 = {ahi[m][0], ahi[m][1]};
            v2f B1 = {sg * ahi[mr][0], sg * ahi[mr][1]};
            c01 = __builtin_amdgcn_wmma_f32_16x16x4_f32(false, A0, false, B1, (short)0, c01, false, false);
            c11 = __builtin_amdgcn_wmma_f32_16x16x4_f32(false, A1, false, B1, (short)0, c11, false, false);
        }
    }
}

template <int L>
__device__ void pair_block(const float* __restrict__ S, const int* __restrict__ alist,
                           int nA, float* __restrict__ outB2, int n_atoms) {
    constexpr int q    = (L == 0) ? 32 : (L == 1) ? 24 : (L == 2) ? 20 : 16;
    constexpr int foff = (L == 0) ? 0 : (L == 1) ? 528 : (L == 2) ? 828 : 1038;
    const float SQRT2 = 1.41421356237309515f;

    const int lane = threadIdx.x & 31;
    const int half = lane >> 4;
    const int row  = lane & 15;
    const int rowh = (16 + row < q) ? (16 + row) : (q - 1);   // address clamp only

    v8f c00 = {}; v8f c01 = {}; v8f c11 = {};

    const int nG   = nA >> 2;
    const int nrem = nA & 3;

    for (int g = 0; g < nG; ++g) {                    // branch-free main loop (EXEC all-ones)
        const int* ap = alist + 4 * g;
        int t0 = ap[0], t1 = ap[1], t2 = ap[2], t3 = ap[3];   // uniform scalar loads
        int at0 = half ? t2 : t0;                     // K slot j=0 for this lane-half
        int at1 = half ? t3 : t1;                     // K slot j=1
        mma_group<L, false>(S, n_atoms, row, rowh, at0, at1, 1.f, 1.f, c00, c01, c11);
    }
    if (nrem > 0) {                                   // wave-uniform branch; masked K tail
        const int* ap = alist + 4 * nG;
        int t0 = ap[0];
        int t1 = ap[(nrem > 1) ? 1 : (nrem - 1)];
        int t2 = ap[(nrem > 2) ? 2 : (nrem - 1)];
        int t3 = ap[nrem - 1];
        int at0 = half ? t2 : t0;
        int at1 = half ? t3 : t1;
        float w0 = ((2 * half + 0) < nrem) ? 1.f : 0.f;
        float w1 = ((2 * half + 1) < nrem) ? 1.f : 0.f;
        mma_group<L, true>(S, n_atoms, row, rowh, at0, at1, w0, w1, c00, c01, c11);
    }

    // scatter upper triangle (row-major triu order: f = M*q - M*(M-1)/2 + (N-M))
#pragma unroll
    for (int r = 0; r < 8; ++r) {                     // tile (0,0)
        int M = r + 8 * half, N = row;
        if (M <= N) {
            int f = foff + M * q - (M * (M - 1)) / 2 + (N - M);
            outB2[f] = c00[r] * ((M == N) ? 1.f : SQRT2);
        }
    }
    if constexpr (q > 16) {
#pragma unroll
        for (int r = 0; r < 8; ++r) {                 // tile (0,1): M<16<=N, always M<N
            int M = r + 8 * half, N = 16 + row;
            if (N < q) {
                int f = foff + M * q - (M * (M - 1)) / 2 + (N - M);
                outB2[f] = c01[r] * SQRT2;
            }
        }
#pragma unroll
        for (int r = 0; r < 8; ++r) {                 // tile (1,1)
            int M = 16 + r + 8 * half, N = 16 + row;
            if (M < q && N < q && M <= N) {
                int f = foff + M * q - (M * (M - 1)) / 2 + (N - M);
                outB2[f] = c11[r] * ((M == N) ? 1.f : SQRT2);
            }
        }
    }
}

__global__ __launch_bounds__(128) void
main_kernel(const float* __restrict__ comp, const float* __restrict__ radial,
            const float* __restrict__ s0, const float* __restrict__ s1,
            const float* __restrict__ s2, const float* __restrict__ s3,
            const int* __restrict__ list, const int* __restrict__ cnt,
            const int* __restrict__ offs, float* __restrict__ out, int n_atoms) {
    const int seg = blockIdx.x;
    const int s   = seg >> 2;
    const int sp  = seg & 3;
    const int nA  = cnt[seg];
    const int* alist = list + offs[seg];
    float* rowp = out + (size_t)s * ROWLEN;

    __builtin_prefetch(alist, 0, 3);

    const int tid = threadIdx.x;
    if (tid < 64) {                               // b1: radial spectrum segment sum (coalesced)
        float acc = 0.f;
        for (int i = 0; i < nA; ++i) acc += radial[(size_t)alist[i] * 64 + tid];
        rowp[B1_OFF + sp * 64 + tid] = acc;
    } else if (tid == 64) {                       // b0: composition segment sum
        float acc = 0.f;
        for (int i = 0; i < nA; ++i) acc += comp[alist[i]];
        rowp[sp] = acc;
    }
    // waves reconverge here; EXEC all-ones entering WMMA region

    float* b2 = rowp + B2_OFF + sp * F2;
    switch (tid >> 5) {
        case 0: pair_block<0>(s0, alist, nA, b2, n_atoms); break;
        case 1: pair_block<1>(s1, alist, nA, b2, n_atoms); break;
        case 2: pair_block<2>(s2, alist, nA, b2, n_atoms); break;
        default: pair_block<3>(s3, alist, nA, b2, n_atoms); break;
    }
}

// ---------------- launcher ----------------

extern "C" void kernel_launch(void* const* d_in, const int* in_sizes, int n_in,
                              void* d_out, int out_size, void* d_ws, size_t ws_size,
                              hipStream_t stream) {
    const float* comp   = (const float*)d_in[0];
    const float* radial = (const float*)d_in[1];
    const float* s0     = (const float*)d_in[2];
    const float* s1     = (const float*)d_in[3];
    const float* s2     = (const float*)d_in[4];
    const float* s3     = (const float*)d_in[5];
    const int*   st     = (const int*)d_in[6];
    const int*   sp     = (const int*)d_in[7];
    const int n_atoms   = in_sizes[6];

    int* cnt  = (int*)d_ws;            // [1024]
    int* offs = cnt + NSEG;            // [1024]
    int* list = offs + NSEG;           // [n_atoms]

    seg_count_kernel<<<NSEG / 128, 128, 0, stream>>>(st, sp, cnt, n_atoms);
    seg_scan_kernel<<<1, NSEG, 0, stream>>>(cnt, offs);
    seg_fill_kernel<<<NSEG / 128, 128, 0, stream>>>(st, sp, offs, list, n_atoms);
    main_kernel<<<NSEG, 128, 0, stream>>>(comp, radial, s0, s1, s2, s3,
                                          list, cnt, offs, (float*)d_out, n_atoms);
}